// LinearConv3D_46265387712811
// MI455X (gfx1250) — compile-verified
//
#include <hip/hip_runtime.h>
#include <hip/hip_bf16.h>
#include <stdint.h>

// ---- problem constants (from reference setup) ----
#define B_   8
#define C_   2
#define D_   96
#define H_   64
#define W_   64
#define F_   8
#define dK_  3
#define h_   60
#define DOUT 94
#define PLANE_HW     (H_ * W_)   // 4096 (input plane)
#define PLANE_HW_OUT (h_ * W_)   // 3840 (output plane)
#define STRIP 20                 // y-strip per block in kMain (3 strips cover h=60)
#define LEAKY 0.2f

// ---- CDNA5 async global->LDS path (guarded; falls back to plain copy) ----
#if defined(__AMDGCN__) && __has_builtin(__builtin_amdgcn_global_load_async_to_lds_b128)
#define ASYNC_OK 1
#define GAS __attribute__((address_space(1)))
#define LAS __attribute__((address_space(3)))
typedef int v4i __attribute__((ext_vector_type(4)));
#else
#define ASYNC_OK 0
#endif

__device__ __forceinline__ void async_copy_b128(const float4* gsrc, float4* ldst) {
#if ASYNC_OK
  __builtin_amdgcn_global_load_async_to_lds_b128((GAS v4i*)(void*)gsrc,
                                                 (LAS v4i*)(void*)ldst,
                                                 /*imm offset*/0, /*cpol*/0);
#else
  *ldst = *gsrc;
#endif
}

__device__ __forceinline__ void wait_async_all() {
#if ASYNC_OK
#if __has_builtin(__builtin_amdgcn_s_wait_asynccnt)
  __builtin_amdgcn_s_wait_asynccnt(0);
#else
  asm volatile("s_wait_asynccnt 0" ::: "memory");
#endif
#endif
}

// =====================================================================
// Kernel 1: BB[f,y,x] = boxsum3x3( 0.25 * sum_{c,k} bias[f,c,k,y,x] )
// grid = F_, block = 256
// =====================================================================
__global__ void kBiasBox(const float* __restrict__ bias, float* __restrict__ BB) {
  __shared__ float bs[PLANE_HW_OUT];  // 15 KB
  const int f = blockIdx.x;
  const int tid = threadIdx.x;

  for (int i = tid; i < PLANE_HW_OUT; i += 256) {
    float s = 0.f;
#pragma unroll
    for (int p = 0; p < C_ * dK_; ++p)
      s += bias[(size_t)(f * C_ * dK_ + p) * PLANE_HW_OUT + i];
    bs[i] = 0.25f * s;
  }
  __syncthreads();

  for (int i = tid; i < PLANE_HW_OUT; i += 256) {
    const int y = i >> 6, x = i & 63;
    float acc = 0.f;
#pragma unroll
    for (int dy = -1; dy <= 1; ++dy) {
      const int yy = y + dy;
      if (yy < 0 || yy >= h_) continue;
      const float* row = bs + yy * W_;
      if (x > 0)   acc += row[x - 1];
      acc += row[x];
      if (x < 63)  acc += row[x + 1];
    }
    BB[(size_t)f * PLANE_HW_OUT + i] = acc;
  }
}

// =====================================================================
// Kernel 2: vs[b,c,z,y,x] = sum_{j=0..4} in[b,c,z,y+j,x]   (y in [0,60))
// grid = B_*C_*D_ (one 64x64 plane per block), block = 256
// =====================================================================
__global__ void kVSum(const float* __restrict__ in, float* __restrict__ vs) {
  __shared__ __align__(16) float pl[PLANE_HW];  // 16 KB
  const int plane = blockIdx.x;                 // (b*C + c)*D + z
  const int tid = threadIdx.x;

  const float4* gin = (const float4*)(in + (size_t)plane * PLANE_HW);
  float4* lp = (float4*)pl;
#pragma unroll
  for (int i = tid; i < PLANE_HW / 4; i += 256)     // 1024 float4
    async_copy_b128(gin + i, lp + i);
  wait_async_all();
  __syncthreads();

  float4* gout = (float4*)(vs + (size_t)plane * PLANE_HW_OUT);
  for (int i = tid; i < PLANE_HW_OUT / 4; i += 256) {  // 960 float4
    const int y = i >> 4;       // row (16 float4 per row)
    const int x4 = i & 15;
    float4 r;
    r.x = r.y = r.z = r.w = 0.f;
#pragma unroll
    for (int j = 0; j < 5; ++j) {
      const float4 v = lp[(y + j) * 16 + x4];
      r.x += v.x; r.y += v.y; r.z += v.z; r.w += v.w;
    }
    gout[i] = r;
  }
}

// =====================================================================
// Kernel 3: for one (b,t) pair and one 20-row y-strip:
//   A[f,y,x] = 0.25 * sum_p vs[b,p,t,y,x] * W[f,p,y,x]   (p = c*3+k, 6 planes)
//   out      = lrelu( boxsum3x3(A) + BB[f,y,x] )
// grid = B_*DOUT*3, block = 256
// =====================================================================
__global__ void kMain(const float* __restrict__ vs,
                      const float* __restrict__ weight,
                      const float* __restrict__ BB,
                      float* __restrict__ out) {
  __shared__ __align__(16) float vst[6 * 22 * W_];  // 33 KB (strip + halo rows)
  __shared__ __align__(16) float at[22 * W_];       // 5.5 KB  (A tile, one f)
  const int tid = threadIdx.x;

  int blk = blockIdx.x;
  const int strip = blk % 3; blk /= 3;
  const int t = blk % DOUT;
  const int b = blk / DOUT;
  const int y0 = strip * STRIP;

  // ---- stage vs rows [y0-1 .. y0+20] for the 6 (c,k) planes; halo rows -> 0 ----
  float4* lv = (float4*)vst;
  for (int i4 = tid; i4 < 6 * 22 * 16; i4 += 256) {   // 2112 float4
    const int p  = i4 / (22 * 16);
    const int r  = (i4 % (22 * 16)) >> 4;
    const int x4 = i4 & 15;
    const int c  = p / 3, k = p % 3;
    const int y  = y0 - 1 + r;
    if (y >= 0 && y < h_) {
      const float4* src = (const float4*)(vs
          + ((size_t)((b * C_ + c) * D_ + (t + k))) * PLANE_HW_OUT + (size_t)y * W_) + x4;
      async_copy_b128(src, lv + i4);
    } else {
      float4 z; z.x = z.y = z.z = z.w = 0.f;
      lv[i4] = z;                                     // ds store, covered by barrier
    }
  }
  wait_async_all();
  __syncthreads();

  const size_t out_b = ((size_t)(b * F_) * DOUT + t) * PLANE_HW_OUT;

  for (int f = 0; f < F_; ++f) {
    // ---- A tile (22 rows incl. halo); rows outside [0,h) are exact zeros ----
    for (int i = tid; i < 22 * W_; i += 256) {        // 1408 elems
      const int r = i >> 6, x = i & 63;
      const int y = y0 - 1 + r;
      float a = 0.f;
      if (y >= 0 && y < h_) {
        const float* wp = weight + (size_t)(f * C_ * dK_) * PLANE_HW_OUT
                        + (size_t)y * W_ + x;
        float acc = 0.f;
#pragma unroll
        for (int p = 0; p < 6; ++p)
          acc = fmaf(vst[p * (22 * W_) + i], wp[(size_t)p * PLANE_HW_OUT], acc);
        a = 0.25f * acc;
      }
      at[i] = a;
    }
    __syncthreads();

    // ---- 3x3 zero-padded box-sum + bias box + leaky-relu + NT store ----
    for (int i = tid; i < STRIP * W_; i += 256) {     // 1280 outputs
      const int iy = i >> 6, x = i & 63;
      const int y = y0 + iy;
      const int r = iy + 1;                           // tile row of center
      float acc = 0.f;
#pragma unroll
      for (int dy = -1; dy <= 1; ++dy) {
        const float* row = at + (r + dy) * W_;
        if (x > 0)   acc += row[x - 1];
        acc += row[x];
        if (x < 63)  acc += row[x + 1];
      }
      const float v = acc + BB[(size_t)f * PLANE_HW_OUT + (size_t)y * W_ + x];
      const float o = v > 0.f ? v : LEAKY * v;
      __builtin_nontemporal_store(o,
          out + out_b + (size_t)f * DOUT * PLANE_HW_OUT + (size_t)y * W_ + x);
    }
    __syncthreads();
  }
}

// =====================================================================
// Host launcher. Workspace layout:
//   vs : B*C*D*h*w floats = 5,898,240  (23.6 MB)
//   BB : F*h*w floats     = 30,720
// total ~23.7 MB  (<< typical ws)
// =====================================================================
extern "C" void kernel_launch(void* const* d_in, const int* in_sizes, int n_in,
                              void* d_out, int out_size, void* d_ws, size_t ws_size,
                              hipStream_t stream) {
  const float* in     = (const float*)d_in[0];
  const float* weight = (const float*)d_in[1];
  const float* bias   = (const float*)d_in[2];
  float* outp = (float*)d_out;

  float* vs = (float*)d_ws;
  float* BB = vs + (size_t)B_ * C_ * D_ * PLANE_HW_OUT;

  (void)in_sizes; (void)n_in; (void)out_size; (void)ws_size;

  kBiasBox<<<F_, 256, 0, stream>>>(bias, BB);
  kVSum<<<B_ * C_ * D_, 256, 0, stream>>>(in, vs);
  kMain<<<B_ * DOUT * 3, 256, 0, stream>>>(vs, weight, BB, outp);
}